// AutoEncoder_46291157516497
// MI455X (gfx1250) — compile-verified
//
#include <hip/hip_runtime.h>

// ---------------- static problem configuration ----------------
#define N_NODES    40960
#define B_GRAPHS   256
#define NPG        160            // nodes per graph
#define EPG        2560           // edges per graph (160*16)
#define N_EDGES    655360
#define IN_DIM     128
#define HID        256
#define LATENT     64
#define HID_DEC    512
#define N_UT       12720
#define N_UT_PAD   12800          // padded to multiple of 256 for WMMA tiles
#define N_MAX      160
#define DEG_CAP    64
#define INV_STD    0.9999950000374997f   // 1/sqrt(1 + 1e-5), BN eval

typedef __attribute__((ext_vector_type(2))) float v2f;
typedef __attribute__((ext_vector_type(8))) float v8f;

// ---------------------------------------------------------------
// Deterministic per-graph CSR build: incoming edge lists in edge order.
// One workgroup per graph; edges staged in LDS; 160 threads each own a node.
// ---------------------------------------------------------------
__global__ void build_csr_kernel(const int* __restrict__ src,
                                 const int* __restrict__ dst,
                                 int* __restrict__ deg,
                                 int* __restrict__ lst) {
    __shared__ int s_src[EPG];
    __shared__ int s_dst[EPG];
    const int g  = blockIdx.x;
    const int e0 = g * EPG;
    for (int e = threadIdx.x; e < EPG; e += blockDim.x) {
        s_src[e] = src[e0 + e];
        s_dst[e] = dst[e0 + e];
    }
    __syncthreads();
    const int t = threadIdx.x;
    if (t < NPG) {
        const int node = g * NPG + t;
        int* row = lst + (size_t)node * DEG_CAP;
        int cnt = 0;
        for (int j = 0; j < EPG; ++j) {
            if (s_dst[j] == node) {
                if (cnt < DEG_CAP) row[cnt] = s_src[j];
                ++cnt;
            }
        }
        deg[node] = cnt < DEG_CAP ? cnt : DEG_CAP;
    }
}

// ---------------------------------------------------------------
// GIN aggregation: z[i] = h[i] + sum_{j in N_in(i)} h[j]  (pure gather)
// One block per node, one thread per feature (d = 128 or 256).
// ---------------------------------------------------------------
__global__ void agg_kernel(const float* __restrict__ h,
                           const int* __restrict__ deg,
                           const int* __restrict__ lst,
                           float* __restrict__ z, int d) {
    const int node = blockIdx.x;
    const int t    = threadIdx.x;
    const int* row = lst + (size_t)node * DEG_CAP;
    float acc = h[(size_t)node * d + t];
    const int dg = deg[node];
    for (int k = 0; k < dg; ++k)
        acc += h[(size_t)row[k] * d + t];
    z[(size_t)node * d + t] = acc;
}

// ---------------------------------------------------------------
// FP32 WMMA GEMM: C[M,N] = A[M,K] @ B[K,N] (+ fused epilogue)
//   EPI 0: leaky(x+b)*scale*INV_STD+shift   (GIN: Linear->LeakyReLU->BN)
//   EPI 1: leaky(x+b)
//   EPI 2: relu(x+b)
//   EPI 3: x+b
// Block = 256 threads (8 waves). Each wave owns a 32(M)x32(N) tile:
// 2 A fragments x 2 B fragments -> 4 V_WMMA_F32_16X16X4_F32 per K-step,
// doubling FLOPs per loaded byte vs a 16x16 tile. 8 waves span 256 columns.
// Layouts per ISA 7.12.2:
//   A 16x4 : lane[0:15] M=lane, vgpr{0,1}=K{0,1}; lane[16:31] K{2,3}
//   B 4x16 : vgpr0 lanes[0:15]=row K0, lanes[16:31]=row K2; vgpr1 = K1/K3
//   C 16x16: c[r] at (M = r + 8*(lane>=16), N = lane&15)
// Requires: M % 32 == 0, N % 256 == 0, K % 4 == 0.
// ---------------------------------------------------------------
template<int EPI>
__device__ __forceinline__ float epilogue(float acc, int n,
                                          const float* __restrict__ bias,
                                          const float* __restrict__ scale,
                                          const float* __restrict__ shift) {
    float v = acc + bias[n];
    if (EPI == 0) { v = v >= 0.0f ? v : 0.2f * v;
                    v = v * (scale[n] * INV_STD) + shift[n]; }
    else if (EPI == 1) { v = v >= 0.0f ? v : 0.2f * v; }
    else if (EPI == 2) { v = v > 0.0f ? v : 0.0f; }
    return v;
}

template<int EPI>
__global__ void gemm_wmma_kernel(const float* __restrict__ A,
                                 const float* __restrict__ B,
                                 float* __restrict__ C,
                                 int M, int N, int K,
                                 const float* __restrict__ bias,
                                 const float* __restrict__ scale,
                                 const float* __restrict__ shift) {
    const int lane = threadIdx.x & 31;
    const int wv   = threadIdx.x >> 5;
    const int lo   = lane & 15;
    const int hi   = lane >> 4;
    const int m0   = blockIdx.y * 32;
    const int n0   = blockIdx.x * 256 + wv * 32;

    // per-lane base pointers; bumped by immediates-friendly strides in loop
    const float* a0p = A + (size_t)(m0 + lo) * K + 2 * hi;        // rows m0..m0+15
    const float* a1p = a0p + (size_t)16 * K;                      // rows m0+16..m0+31
    const float* bp  = B + (size_t)(2 * hi) * N + n0 + lo;        // rows 2hi, 2hi+1

    v8f acc00 = {}, acc01 = {}, acc10 = {}, acc11 = {};
    for (int k0 = 0; k0 < K; k0 += 4) {
        v2f a0 = *(const v2f*)a0p;            // A[m][k0+2hi], A[m][k0+2hi+1]
        v2f a1 = *(const v2f*)a1p;
        v2f b0, b1;
        b0.x = bp[0];   b0.y = bp[N];         // rows k0+2hi, k0+2hi+1
        b1.x = bp[16];  b1.y = bp[N + 16];
        a0p += 4; a1p += 4; bp += (size_t)4 * N;
        acc00 = __builtin_amdgcn_wmma_f32_16x16x4_f32(
                    false, a0, false, b0, (short)0, acc00, false, false);
        acc01 = __builtin_amdgcn_wmma_f32_16x16x4_f32(
                    false, a0, false, b1, (short)0, acc01, false, false);
        acc10 = __builtin_amdgcn_wmma_f32_16x16x4_f32(
                    false, a1, false, b0, (short)0, acc10, false, false);
        acc11 = __builtin_amdgcn_wmma_f32_16x16x4_f32(
                    false, a1, false, b1, (short)0, acc11, false, false);
    }

    const int na = n0 + lo;
    const int nb = n0 + 16 + lo;
#pragma unroll
    for (int r = 0; r < 8; ++r) {
        const int ma = m0 + r + 8 * hi;
        const int mb = ma + 16;
        C[(size_t)ma * N + na] = epilogue<EPI>(acc00[r], na, bias, scale, shift);
        C[(size_t)ma * N + nb] = epilogue<EPI>(acc01[r], nb, bias, scale, shift);
        C[(size_t)mb * N + na] = epilogue<EPI>(acc10[r], na, bias, scale, shift);
        C[(size_t)mb * N + nb] = epilogue<EPI>(acc11[r], nb, bias, scale, shift);
    }
}

// ---------------------------------------------------------------
// global_add_pool: pooled[g][c] = sum over 160 consecutive node rows
// ---------------------------------------------------------------
__global__ void pool_kernel(const float* __restrict__ h,
                            float* __restrict__ pooled) {
    const int g = blockIdx.x;
    const int c = threadIdx.x;
    const float* base = h + (size_t)g * NPG * HID + c;
    float acc = 0.0f;
    for (int i = 0; i < NPG; ++i) acc += base[(size_t)i * HID];
    pooled[g * HID + c] = acc;
}

// pooled-BN + fc to latent: lat[g][j] = sum_c (p*g_bn*INV_STD+b_bn) * W + b
__global__ void fc_kernel(const float* __restrict__ pooled,
                          const float* __restrict__ bn_g,
                          const float* __restrict__ bn_b,
                          const float* __restrict__ W,
                          const float* __restrict__ b,
                          float* __restrict__ lat) {
    const int id = blockIdx.x * blockDim.x + threadIdx.x;   // 256*64
    const int g = id >> 6, j = id & 63;
    float acc = 0.0f;
    for (int c = 0; c < HID; ++c) {
        const float p = pooled[g * HID + c] * (bn_g[c] * INV_STD) + bn_b[c];
        acc += p * W[c * LATENT + j];
    }
    lat[id] = acc + b[j];
}

// decoder layer 0 (K=64, tiny): relu(lat @ W + b)
__global__ void dec1_kernel(const float* __restrict__ lat,
                            const float* __restrict__ W,
                            const float* __restrict__ b,
                            float* __restrict__ out) {
    const int id = blockIdx.x * blockDim.x + threadIdx.x;   // 256*512
    const int g = id >> 9, j = id & 511;
    float acc = 0.0f;
    for (int c = 0; c < LATENT; ++c)
        acc += lat[g * LATENT + c] * W[c * HID_DEC + j];
    const float v = acc + b[j];
    out[id] = v > 0.0f ? v : 0.0f;
}

// pad W3 (512 x 12720) -> (512 x 12800) with zero columns (into d_out scratch)
__global__ void pad_w3_kernel(const float* __restrict__ W3,
                              float* __restrict__ W3p) {
    const int id = blockIdx.x * blockDim.x + threadIdx.x;   // 512*12800
    const int k = id / N_UT_PAD;
    const int n = id - k * N_UT_PAD;
    W3p[id] = (n < N_UT) ? W3[(size_t)k * N_UT + n] : 0.0f;
}

__global__ void pad_b3_kernel(const float* __restrict__ b3,
                              float* __restrict__ b3p) {
    const int n = blockIdx.x * blockDim.x + threadIdx.x;
    if (n < N_UT_PAD) b3p[n] = (n < N_UT) ? b3[n] : 0.0f;
}

// scatter upper-tri logits into symmetric adjacency, diag = -10
__global__ void adj_kernel(const float* __restrict__ logits,
                           float* __restrict__ out) {
    const int id = blockIdx.x * blockDim.x + threadIdx.x;   // 256*160*160
    const int g   = id / (N_MAX * N_MAX);
    const int rem = id - g * (N_MAX * N_MAX);
    const int i = rem / N_MAX;
    const int j = rem - i * N_MAX;
    float v;
    if (i == j) {
        v = -10.0f;
    } else {
        const int a = i < j ? i : j;
        const int b = i < j ? j : i;
        const int idx = a * (2 * N_MAX - 1 - a) / 2 + (b - a - 1);
        v = logits[(size_t)g * N_UT_PAD + idx];
    }
    out[id] = v;
}

// ---------------------------------------------------------------
extern "C" void kernel_launch(void* const* d_in, const int* in_sizes, int n_in,
                              void* d_out, int out_size, void* d_ws, size_t ws_size,
                              hipStream_t stream) {
    (void)in_sizes; (void)n_in; (void)out_size; (void)ws_size;

    const float* x   = (const float*)d_in[0];
    const int*   ei  = (const int*)d_in[1];          // int32 (jax default x32)
    const int*   src = ei;
    const int*   dst = ei + N_EDGES;
    // d_in[2] = batch : implicit (160 consecutive nodes per graph)

    // encoder params: 4 layers x {W1,b1,g,bt,W2,b2} starting at index 3
    const float* eW1[4]; const float* eb1[4]; const float* eg[4];
    const float* ebt[4]; const float* eW2[4]; const float* eb2[4];
    int p = 3;
    for (int l = 0; l < 4; ++l) {
        eW1[l] = (const float*)d_in[p++]; eb1[l] = (const float*)d_in[p++];
        eg[l]  = (const float*)d_in[p++]; ebt[l] = (const float*)d_in[p++];
        eW2[l] = (const float*)d_in[p++]; eb2[l] = (const float*)d_in[p++];
    }
    const float* bn_g = (const float*)d_in[p++];     // 27
    const float* bn_b = (const float*)d_in[p++];     // 28
    const float* fc_W = (const float*)d_in[p++];     // 29
    const float* fc_b = (const float*)d_in[p++];     // 30
    const float* dW0  = (const float*)d_in[p++];     // 31  64x512
    const float* db0  = (const float*)d_in[p++];     // 32
    const float* dW1  = (const float*)d_in[p++];     // 33  512x512
    const float* db1  = (const float*)d_in[p++];     // 34
    const float* dW2  = (const float*)d_in[p++];     // 35  512x12720
    const float* db2  = (const float*)d_in[p++];     // 36

    // ---- workspace layout (~152 MB) ----
    float* ws = (float*)d_ws;
    size_t off = 0;
    float* bufA = ws + off; off += (size_t)N_NODES * HID;       // h
    float* bufB = ws + off; off += (size_t)N_NODES * HID;       // z = h + agg
    float* bufC = ws + off; off += (size_t)N_NODES * HID;       // after lin1+BN
    int*   deg  = (int*)(ws + off); off += N_NODES;
    int*   lst  = (int*)(ws + off); off += (size_t)N_NODES * DEG_CAP;
    float* pooled = ws + off; off += (size_t)B_GRAPHS * HID;
    float* lat    = ws + off; off += (size_t)B_GRAPHS * LATENT;
    float* dbuf1  = ws + off; off += (size_t)B_GRAPHS * HID_DEC;
    float* dbuf2  = ws + off; off += (size_t)B_GRAPHS * HID_DEC;
    float* b3pad  = ws + off; off += N_UT_PAD;
    float* logits = ws + off; off += (size_t)B_GRAPHS * N_UT_PAD;

    // padded W3 scratch: exactly out_size floats (512*12800 == 256*160*160)
    float* W3pad = (float*)d_out;

    // 1) deterministic CSR (rebuilt every call; no cached state)
    build_csr_kernel<<<B_GRAPHS, 256, 0, stream>>>(src, dst, deg, lst);

    // 2) pad decoder final weight/bias for 256-wide WMMA tiling
    pad_w3_kernel<<<(HID_DEC * N_UT_PAD) / 256, 256, 0, stream>>>(dW2, W3pad);
    pad_b3_kernel<<<(N_UT_PAD + 255) / 256, 256, 0, stream>>>(db2, b3pad);

    // 3) GIN encoder: 4 layers
    for (int l = 0; l < 4; ++l) {
        const float* hin = (l == 0) ? x : bufA;
        const int d = (l == 0) ? IN_DIM : HID;
        agg_kernel<<<N_NODES, d, 0, stream>>>(hin, deg, lst, bufB, d);
        // bufC = BN(leaky(bufB @ W1 + b1))
        gemm_wmma_kernel<0><<<dim3(1, N_NODES / 32), 256, 0, stream>>>(
            bufB, eW1[l], bufC, N_NODES, HID, d, eb1[l], eg[l], ebt[l]);
        // bufA = leaky(bufC @ W2 + b2)
        gemm_wmma_kernel<1><<<dim3(1, N_NODES / 32), 256, 0, stream>>>(
            bufC, eW2[l], bufA, N_NODES, HID, HID, eb2[l], nullptr, nullptr);
    }

    // 4) pool + BN + fc -> latent
    pool_kernel<<<B_GRAPHS, HID, 0, stream>>>(bufA, pooled);
    fc_kernel<<<(B_GRAPHS * LATENT) / 256, 256, 0, stream>>>(
        pooled, bn_g, bn_b, fc_W, fc_b, lat);

    // 5) decoder
    dec1_kernel<<<(B_GRAPHS * HID_DEC) / 256, 256, 0, stream>>>(
        lat, dW0, db0, dbuf1);
    gemm_wmma_kernel<2><<<dim3(HID_DEC / 256, B_GRAPHS / 32), 256, 0, stream>>>(
        dbuf1, dW1, dbuf2, B_GRAPHS, HID_DEC, HID_DEC, db1, nullptr, nullptr);
    gemm_wmma_kernel<3><<<dim3(N_UT_PAD / 256, B_GRAPHS / 32), 256, 0, stream>>>(
        dbuf2, W3pad, logits, B_GRAPHS, N_UT_PAD, HID_DEC, b3pad, nullptr, nullptr);

    // 6) symmetric adjacency with -10 diagonal (overwrites d_out scratch)
    adj_kernel<<<(B_GRAPHS * N_MAX * N_MAX) / 256, 256, 0, stream>>>(logits, (float*)d_out);
}